// MultiheadAttention_1992864825946
// MI455X (gfx1250) — compile-verified
//
#include <hip/hip_runtime.h>
#include <hip/hip_bf16.h>

#define S 2048
#define D 64
#define BH 16
#define NEG_FILL (-4294967295.0f)   // -(2^32)+1, matches reference masked_fill

typedef float v2f __attribute__((ext_vector_type(2)));
typedef float v8f __attribute__((ext_vector_type(8)));

extern __shared__ char smem_raw[];

__global__ __launch_bounds__(256) void mha_fwd_kernel(
    const float* __restrict__ q,
    const float* __restrict__ k,
    const float* __restrict__ v,
    const unsigned char* __restrict__ mask,   // bool, 1 byte/elem
    const float* __restrict__ qmask,
    float* __restrict__ out,                  // [BH,S,D]
    float* __restrict__ attn_out)             // [BH,S,S]
{
    float* sc   = (float*)smem_raw;      // [16][S] score/attn strip: 128 KB
    float* part = sc + 16 * S;           // [4][256] phase-C partials: 4 KB

    const int bh   = blockIdx.y;
    const int q0   = blockIdx.x * 16;
    const int tid  = threadIdx.x;
    const int wave = tid >> 5;
    const int lane = tid & 31;
    const int lrow = lane & 15;          // row/col-within-tile this lane owns
    const int hi   = lane >> 4;          // K sub-offset selector for A/B

    // ---------------- Phase A: scores = Q K^T / sqrt(D), mask, -> LDS -------
    // A-matrix (16x4 f32): lane holds Q[lrow][4t + 2*hi + {0,1}] -> float2.
    const float* qrow = q + ((size_t)(bh * S + q0 + lrow)) * D;
    v2f a[16];
#pragma unroll
    for (int t = 0; t < 16; ++t)
        a[t] = *(const v2f*)(qrow + 4 * t + 2 * hi);

    const unsigned char* mbase = mask + (size_t)bh * S * S;

    for (int tile = wave; tile < S / 16; tile += 8) {
        const int kcol0 = tile * 16;
        // B-matrix (4x16 f32): B[k][n] = Key[n][k]; same float2 pattern as A.
        const float* krow = k + ((size_t)(bh * S + kcol0 + lrow)) * D;
        v8f c = {};
#pragma unroll
        for (int t = 0; t < 16; ++t) {
            v2f b = *(const v2f*)(krow + 4 * t + 2 * hi);
            c = __builtin_amdgcn_wmma_f32_16x16x4_f32(
                    false, a[t], false, b, (short)0, c, false, false);
        }
        // C layout: VGPR r -> M = r + 8*hi, N = lrow. Scale + mask + stash.
        const int ncol = kcol0 + lrow;
#pragma unroll
        for (int r = 0; r < 8; ++r) {
            const int m = r + 8 * hi;
            const unsigned char mk = mbase[(size_t)(q0 + m) * S + ncol];
            sc[m * S + ncol] = mk ? NEG_FILL : c[r] * 0.125f;
        }
    }
    __syncthreads();

    // ---------------- Phase B: row softmax * query_mask, write attn --------
    for (int rr = 0; rr < 2; ++rr) {
        const int row = wave * 2 + rr;
        float* srow = sc + row * S;

        float mx = -3.4e38f;
        for (int j = lane; j < S; j += 32) mx = fmaxf(mx, srow[j]);
#pragma unroll
        for (int o = 16; o > 0; o >>= 1) mx = fmaxf(mx, __shfl_xor(mx, o, 32));

        float sum = 0.0f;
        for (int j = lane; j < S; j += 32) {
            const float p = __expf(srow[j] - mx);
            srow[j] = p;
            sum += p;
        }
#pragma unroll
        for (int o = 16; o > 0; o >>= 1) sum += __shfl_xor(sum, o, 32);

        const float factor = qmask[bh * S + q0 + row] / sum;
        float* arow = attn_out + ((size_t)(bh * S + q0 + row)) * S;
        for (int j = lane; j < S; j += 32) {
            const float val = srow[j] * factor;
            srow[j] = val;       // reused by Phase C
            arow[j] = val;       // coalesced 128B stores (dominant traffic)
        }
    }
    __syncthreads();

    // ---------------- Phase C: out = P @ V  (16 x 2048 @ 2048 x 64) --------
    // wave -> (ntile = wave&3 over N=64, khalf = wave>>2 over K=2048)
    const int ntile = wave & 3;
    const int khalf = wave >> 2;
    const int ncol  = ntile * 16 + lrow;
    const float* vcol = v + (size_t)bh * S * D + ncol;   // V[k][ncol] @ +k*D

    v8f acc = {};
    const int kbeg = khalf * (S / 2);
    for (int k0 = kbeg; k0 < kbeg + S / 2; k0 += 4) {
        // A from LDS: P[lrow][k0 + 2*hi + {0,1}] (8B aligned -> ds_load_b64)
        v2f av = *(const v2f*)(sc + lrow * S + k0 + 2 * hi);
        v2f bv;
        bv.x = vcol[(size_t)(k0 + 2 * hi) * D];
        bv.y = vcol[(size_t)(k0 + 1 + 2 * hi) * D];
        acc = __builtin_amdgcn_wmma_f32_16x16x4_f32(
                  false, av, false, bv, (short)0, acc, false, false);
    }

    // deterministic cross-wave pair reduction through LDS
    if (khalf == 1) {
#pragma unroll
        for (int r = 0; r < 8; ++r)
            part[ntile * 256 + r * 32 + lane] = acc[r];
    }
    __syncthreads();
    if (khalf == 0) {
#pragma unroll
        for (int r = 0; r < 8; ++r) {
            const float val = acc[r] + part[ntile * 256 + r * 32 + lane];
            const int m = r + 8 * hi;
            out[((size_t)(bh * S + q0 + m)) * D + ncol] = val;
        }
    }
}

extern "C" void kernel_launch(void* const* d_in, const int* in_sizes, int n_in,
                              void* d_out, int out_size, void* d_ws, size_t ws_size,
                              hipStream_t stream) {
    // setup_inputs order: key, value, query, mask, query_mask
    const float*         key   = (const float*)d_in[0];
    const float*         value = (const float*)d_in[1];
    const float*         query = (const float*)d_in[2];
    const unsigned char* mask  = (const unsigned char*)d_in[3];
    const float*         qmask = (const float*)d_in[4];

    float* out      = (float*)d_out;                    // result [16,2048,64]
    float* attn_out = out + (size_t)BH * S * D;         // attention [16,2048,2048]

    const size_t smem = (size_t)(16 * S + 4 * 256) * sizeof(float); // ~132 KB
    mha_fwd_kernel<<<dim3(S / 16, BH), 256, smem, stream>>>(
        query, key, value, mask, qmask, out, attn_out);
}